// EnsRec_7739531067850
// MI455X (gfx1250) — compile-verified
//
#include <hip/hip_runtime.h>
#include <hip/hip_bf16.h>
#include <math.h>

// ---- problem constants (match reference) ----
#define BB      512
#define KK      8
#define LL      50
#define DLLM    768
#define HH      128
#define N_ITEM  100000
#define DIV_TRADEOFF 0.1f

typedef __attribute__((ext_vector_type(2))) float v2f;
typedef __attribute__((ext_vector_type(8))) float v8f;

#define LDS_STRIDE 772   // 768 + 4 pad: lane m hits bank (4m+k)%64 -> conflict-free A reads

// =====================================================================
// Kernel A: fused masked-gather + time-weighted sum + f32 WMMA GEMM
//   S[row, d] = sum_l tw[l] * item_emb[ids[row,l], d] * valid   (LDS)
//   E[row, h] = S[row,:] @ W_proj[:,h] + (sum_l tw[l]) * b_proj[h]
// One block = 16 rows (rows = b*K + k, 4096 total -> 256 blocks).
// 256 threads = 8 waves; wave w computes N-tile w via V_WMMA_F32_16X16X4_F32.
// =====================================================================
__global__ __launch_bounds__(256)
void ensrec_gather_gemm(const int* __restrict__ preds,       // [B,K,L]
                        const float* __restrict__ item_emb,  // [N_ITEM+1, 768]
                        const float* __restrict__ Wproj,     // [768, 128]
                        const float* __restrict__ bproj,     // [128]
                        float* __restrict__ E)               // [4096, 128] (ws)
{
    __shared__ float tw[64];
    __shared__ float Sld[16][LDS_STRIDE];

    const int t    = threadIdx.x;          // 0..255
    const int row0 = blockIdx.x * 16;

    if (t < LL) tw[t] = 1.0f / log2f((float)t + 2.0f);
    __syncthreads();

    // ---- gather phase: thread t owns d = {t, t+256, t+512} of each row ----
    for (int r = 0; r < 16; ++r) {
        const int* idp = preds + (size_t)(row0 + r) * LL;
        float a0 = 0.f, a1 = 0.f, a2 = 0.f;
        #pragma unroll 2
        for (int l = 0; l < LL; ++l) {
            int id = idp[l];                       // uniform across block -> scalar load
            if (id > 0 && id <= N_ITEM) {
                const float* e = item_emb + (size_t)id * DLLM;
                float w = tw[l];
                a0 = fmaf(w, e[t],       a0);
                a1 = fmaf(w, e[t + 256], a1);
                a2 = fmaf(w, e[t + 512], a2);
            }
        }
        Sld[r][t]       = a0;
        Sld[r][t + 256] = a1;
        Sld[r][t + 512] = a2;
    }
    __syncthreads();

    // ---- WMMA phase: wave w -> output cols [16w, 16w+16) ----
    const int wave = t >> 5;
    const int lane = t & 31;
    const int m    = lane & 15;              // A row within tile
    const int kh   = (lane >> 4) << 1;       // K sub-offset: 0 (lanes 0-15) / 2 (lanes 16-31)
    const int n    = (wave << 4) + (lane & 15); // absolute output column

    v8f acc = {};
    for (int k0 = 0; k0 < DLLM; k0 += 4) {
        v2f a, b;
        a.x = Sld[m][k0 + kh];
        a.y = Sld[m][k0 + kh + 1];
        const float* wp = Wproj + (size_t)(k0 + kh) * HH + n;
        b.x = wp[0];
        b.y = wp[HH];
        acc = __builtin_amdgcn_wmma_f32_16x16x4_f32(
                  false, a, false, b, (short)0, acc, false, false);
    }

    // bias folded with total time weight
    float stw = 0.f;
    #pragma unroll
    for (int l = 0; l < LL; ++l) stw += tw[l];
    const float bias = stw * bproj[n];

    // D layout: VGPR r, lanes 0-15 -> M=r; lanes 16-31 -> M=r+8; N = lane&15
    const int mh = (lane >> 4) << 3;
    #pragma unroll
    for (int r = 0; r < 8; ++r) {
        int row = row0 + r + mh;
        E[(size_t)row * HH + n] = acc[r] + bias;
    }
}

// =====================================================================
// Kernel B: per-batch-row loss (one wave32 per b)
// =====================================================================
__device__ __forceinline__ float wred32(float v) {
    #pragma unroll
    for (int m = 16; m > 0; m >>= 1) v += __shfl_xor(v, m);
    return v;
}

__global__ __launch_bounds__(32)
void ensrec_loss(const int* __restrict__ user_id,
                 const float* __restrict__ pref_in,   // [B,128]
                 const float* __restrict__ pos_label, // [B,8]
                 const float* __restrict__ neg_label, // [B,8]
                 const float* __restrict__ uemb,      // [50000,128]
                 const float* __restrict__ E,         // [4096,128] (ws)
                 float* __restrict__ partial)         // [512] (ws)
{
    const int b    = blockIdx.x;
    const int lane = threadIdx.x;

    const float* Eb = E + (size_t)b * KK * HH;
    const int uid = user_id[b];

    float pref[4], e[KK][4];
    #pragma unroll
    for (int j = 0; j < 4; ++j) {
        int h = lane + 32 * j;
        pref[j] = pref_in[(size_t)b * HH + h] + uemb[(size_t)uid * HH + h];
    }
    #pragma unroll
    for (int k = 0; k < KK; ++k)
        #pragma unroll
        for (int j = 0; j < 4; ++j)
            e[k][j] = Eb[k * HH + lane + 32 * j];

    // attention logits + softmax over K (all lanes hold full result)
    float wg[KK];
    #pragma unroll
    for (int k = 0; k < KK; ++k) {
        float p = 0.f;
        #pragma unroll
        for (int j = 0; j < 4; ++j) p = fmaf(e[k][j], pref[j], p);
        wg[k] = wred32(p);
    }
    float mx = wg[0];
    #pragma unroll
    for (int k = 1; k < KK; ++k) mx = fmaxf(mx, wg[k]);
    float den = 0.f;
    #pragma unroll
    for (int k = 0; k < KK; ++k) { wg[k] = expf(wg[k] - mx); den += wg[k]; }
    float rden = 1.0f / den;
    #pragma unroll
    for (int k = 0; k < KK; ++k) wg[k] *= rden;

    // BPR loss
    float pos = 0.f, neg = 0.f;
    #pragma unroll
    for (int k = 0; k < KK; ++k) {
        pos = fmaf(wg[k], pos_label[b * KK + k], pos);
        neg = fmaf(wg[k], neg_label[b * KK + k], neg);
    }
    float x  = pos - neg;
    float ls = fminf(x, 0.f) - log1pf(expf(-fabsf(x)));   // log_sigmoid(x)
    float loss_b = -ls;

    // diversity: sum_{i!=j} dot(e_i,e_j)^2 * (w_i + w_j)  (use symmetry: 2 * sum_{i<j})
    float term = 0.f;
    #pragma unroll
    for (int i = 0; i < KK; ++i) {
        #pragma unroll
        for (int jj = i + 1; jj < KK; ++jj) {
            float p = 0.f;
            #pragma unroll
            for (int j = 0; j < 4; ++j) p = fmaf(e[i][j], e[jj][j], p);
            float d = wred32(p);
            float g = d * d;           // gram = dot^2
            term = fmaf(g, wg[i] + wg[jj], term);
        }
    }
    term *= 2.0f;

    float out = loss_b + DIV_TRADEOFF * term / (float)(BB * KK * KK);
    if (lane == 0) partial[b] = out;
}

// =====================================================================
// Kernel C: reduce 512 partials -> scalar
// =====================================================================
__global__ __launch_bounds__(512)
void ensrec_reduce(const float* __restrict__ partial, float* __restrict__ out)
{
    __shared__ float s[512];
    int t = threadIdx.x;
    s[t] = partial[t];
    __syncthreads();
    for (int st = 256; st > 0; st >>= 1) {
        if (t < st) s[t] += s[t + st];
        __syncthreads();
    }
    if (t == 0) out[0] = s[0];
}

// =====================================================================
extern "C" void kernel_launch(void* const* d_in, const int* in_sizes, int n_in,
                              void* d_out, int out_size, void* d_ws, size_t ws_size,
                              hipStream_t stream)
{
    (void)in_sizes; (void)n_in; (void)out_size; (void)ws_size;

    const int*   user_id   = (const int*)  d_in[0];
    const int*   preds     = (const int*)  d_in[1];
    const float* pref_in   = (const float*)d_in[2];
    const float* pos_label = (const float*)d_in[3];
    const float* neg_label = (const float*)d_in[4];
    const float* uemb      = (const float*)d_in[5];
    const float* item_emb  = (const float*)d_in[6];
    const float* Wproj     = (const float*)d_in[7];
    const float* bproj     = (const float*)d_in[8];

    float* E       = (float*)d_ws;                                   // 4096*128 f32 = 2 MB
    float* partial = (float*)((char*)d_ws + (size_t)BB * KK * HH * 4); // +512 f32

    // Stage 1: fused gather + f32 WMMA projection (dominant: ~630 MB HBM gather)
    ensrec_gather_gemm<<<(BB * KK) / 16, 256, 0, stream>>>(preds, item_emb, Wproj, bproj, E);
    // Stage 2: per-row attention/softmax/BPR + diversity term
    ensrec_loss<<<BB, 32, 0, stream>>>(user_id, pref_in, pos_label, neg_label, uemb, E, partial);
    // Stage 3: scalar reduction
    ensrec_reduce<<<1, 512, 0, stream>>>(partial, (float*)d_out);
}